// PatchQuantumGenerator_35364760715298
// MI455X (gfx1250) — compile-verified
//
#include <hip/hip_runtime.h>
#include <hip/hip_bf16.h>
#include <math.h>

#define NQ    12
#define QD    3
#define NGEN  2
#define NPAIR (NQ / 2)      // 6 two-qubit fused gates per layer
#define BATCH 4096
#define DIMQ  4096          // 2^NQ
#define TEMP  5.0f

typedef __attribute__((ext_vector_type(2))) float v2f;
typedef __attribute__((ext_vector_type(8))) float v8f;

// ---------------------------------------------------------------------------
// Compile-time tracking of the CNOT-ring permutation.
// Qubit j <-> bit (NQ-1-j) of the flat index (C-order reshape).
// CNOT(c->t) maps l -> l ^ (((l>>pc)&1)<<pt); tracking phi (logical->physical)
// without moving data updates columns: col[pc] ^= col[pt]  (phi' = phi∘C).
// Layer-d gates see phi = ring^d; the final readout sees ring^3.
// ---------------------------------------------------------------------------
struct Perm { unsigned col[NQ]; };

constexpr Perm ring_pow(int n) {
  Perm p{};
  for (int j = 0; j < NQ; ++j) p.col[j] = 1u << j;
  for (int it = 0; it < n; ++it)
    for (int y = 0; y < NQ; ++y) {
      int pc = NQ - 1 - y;                 // control bit
      int pt = NQ - 1 - ((y + 1) % NQ);    // target bit
      p.col[pc] ^= p.col[pt];
    }
  return p;
}

// ---------------------------------------------------------------------------
// Fused single-qubit matrix U = RY(w2)*RX(w1)*RZ(w0).
// ---------------------------------------------------------------------------
__device__ __forceinline__ void fused_u(const float* __restrict__ wb,
                                        float2 u[4]) {
  float a = wb[0], b = wb[1], cgam = wb[2];
  float sa, ca, sb, cb, sc, cc;
  sincosf(0.5f * a, &sa, &ca);
  sincosf(0.5f * b, &sb, &cb);
  sincosf(0.5f * cgam, &sc, &cc);
  // M1 = RX(b) * RZ(a);  RZ = diag(e^{-ia/2}, e^{+ia/2}), RX offdiag = -i*sb
  float2 m00 = make_float2( cb * ca, -cb * sa);
  float2 m01 = make_float2( sb * sa, -sb * ca);
  float2 m10 = make_float2(-sb * sa, -sb * ca);
  float2 m11 = make_float2( cb * ca,  cb * sa);
  // U = RY(c) * M1  (RY is real [[cc,-sc],[sc,cc]])
  u[0] = make_float2(cc * m00.x - sc * m10.x, cc * m00.y - sc * m10.y);
  u[1] = make_float2(cc * m01.x - sc * m11.x, cc * m01.y - sc * m11.y);
  u[2] = make_float2(sc * m00.x + cc * m10.x, sc * m00.y + cc * m10.y);
  u[3] = make_float2(sc * m01.x + cc * m11.x, sc * m01.y + cc * m11.y);
}

// ---------------------------------------------------------------------------
// Kernel 1: build 4x4 complex two-qubit gates K = U(q=2Q) ⊗ U(q=2Q+1) per
// (gen, depth, pair). 36 matrices * 16 cplx -> d_ws (576 float2 = 4608 B).
// Row/col index = (qubit2Q_bit)*2 + (qubit2Q+1_bit).
// ---------------------------------------------------------------------------
__global__ void fuse_pairs_kernel(const float* __restrict__ qp,
                                  float2* __restrict__ U) {
  int idx = blockIdx.x * blockDim.x + threadIdx.x;
  if (idx >= NGEN * QD * NPAIR) return;
  int g = idx / (QD * NPAIR);
  int r = idx % (QD * NPAIR);
  int d = r / NPAIR;
  int q = r % NPAIR;
  const float* base = qp + g * (QD * NQ * 3) + d * (NQ * 3);
  float2 u1[4], u2[4];
  fused_u(base + 3 * (2 * q),     u1);   // qubit 2q     (hi bit of pair)
  fused_u(base + 3 * (2 * q + 1), u2);   // qubit 2q + 1 (lo bit of pair)
  float2* dst = U + (size_t)idx * 16;
#pragma unroll
  for (int ap = 0; ap < 2; ++ap)
#pragma unroll
    for (int bp = 0; bp < 2; ++bp)
#pragma unroll
      for (int aa = 0; aa < 2; ++aa)
#pragma unroll
        for (int bb = 0; bb < 2; ++bb) {
          float2 x1 = u1[ap * 2 + aa];
          float2 x2 = u2[bp * 2 + bb];
          dst[(ap * 2 + bp) * 4 + (aa * 2 + bb)] =
              make_float2(x1.x * x2.x - x1.y * x2.y,
                          x1.x * x2.y + x1.y * x2.x);
        }
}

// ---------------------------------------------------------------------------
// One fused two-qubit gate at layer D on qubits (2Q, 2Q+1), applied through
// the compile-time permutation ring^D. All masks are immediates after unroll.
// ---------------------------------------------------------------------------
__device__ __forceinline__ void cfma(float2& n, float2 M, float2 a) {
  n.x += M.x * a.x - M.y * a.y;
  n.y += M.x * a.y + M.y * a.x;
}

template <int D, int Q>
__device__ __forceinline__ void apply_pair(float2* sAmp,
                                           const float2* __restrict__ sUd,
                                           int tid) {
  constexpr Perm P = ring_pow(D);
  constexpr int p1 = NQ - 1 - 2 * Q;       // logical bit of qubit 2Q  (hi)
  constexpr int p2 = p1 - 1;               // logical bit of qubit 2Q+1 (lo)
  constexpr unsigned m1 = P.col[p1];
  constexpr unsigned m2 = P.col[p2];

  float2 M[16];
#pragma unroll
  for (int i = 0; i < 16; ++i) M[i] = sUd[Q * 16 + i];

  // h enumerates the 1024 logical quads: h = s*256 + tid.
  // Logical l0 = insert two 0 bits at p2,p1 (adjacent); physical = xor-fold.
  unsigned base = 0;
#pragma unroll
  for (int k = 0; k < 8; ++k) {
    unsigned c = P.col[(k < p2) ? k : k + 2];
    base ^= ((tid >> k) & 1) ? c : 0u;
  }
  constexpr unsigned c8 = P.col[(8 < p2) ? 8 : 10];
  constexpr unsigned c9 = P.col[(9 < p2) ? 9 : 11];

#pragma unroll 1
  for (int s = 0; s < 4; ++s) {
    unsigned i00 = base ^ ((s & 1) ? c8 : 0u) ^ ((s & 2) ? c9 : 0u);
    unsigned i01 = i00 ^ m2;
    unsigned i10 = i00 ^ m1;
    unsigned i11 = i10 ^ m2;
    float2 a0 = sAmp[i00];
    float2 a1 = sAmp[i01];
    float2 a2 = sAmp[i10];
    float2 a3 = sAmp[i11];
    float2 n0 = make_float2(0.f, 0.f), n1 = n0, n2 = n0, n3 = n0;
    cfma(n0, M[0],  a0); cfma(n0, M[1],  a1); cfma(n0, M[2],  a2); cfma(n0, M[3],  a3);
    cfma(n1, M[4],  a0); cfma(n1, M[5],  a1); cfma(n1, M[6],  a2); cfma(n1, M[7],  a3);
    cfma(n2, M[8],  a0); cfma(n2, M[9],  a1); cfma(n2, M[10], a2); cfma(n2, M[11], a3);
    cfma(n3, M[12], a0); cfma(n3, M[13], a1); cfma(n3, M[14], a2); cfma(n3, M[15], a3);
    sAmp[i00] = n0;
    sAmp[i01] = n1;
    sAmp[i10] = n2;
    sAmp[i11] = n3;
  }
  __syncthreads();
}

template <int D>
__device__ __forceinline__ void apply_layer(float2* sAmp,
                                            const float2* __restrict__ sUd,
                                            int tid) {
  apply_pair<D, 0>(sAmp, sUd, tid);
  apply_pair<D, 1>(sAmp, sUd, tid);
  apply_pair<D, 2>(sAmp, sUd, tid);
  apply_pair<D, 3>(sAmp, sUd, tid);
  apply_pair<D, 4>(sAmp, sUd, tid);
  apply_pair<D, 5>(sAmp, sUd, tid);
}

// ---------------------------------------------------------------------------
// Kernel 2: 12-qubit state-vector simulation, one block per (sample, gen).
// Interleaved complex state in LDS (ds_*_b64), CNOTs free via compile-time
// permutation, two-qubit fused gates: 18 sweeps + init + readout.
// ---------------------------------------------------------------------------
__global__ __launch_bounds__(256)
void circuit_kernel(const float* __restrict__ x,
                    const float2* __restrict__ Uall,
                    float* __restrict__ out) {
  const int b   = blockIdx.x;
  const int g   = blockIdx.y;
  const int tid = threadIdx.x;

  __shared__ float2 sAmp[DIMQ];                 // 32 KB interleaved re/im
  __shared__ float2 sU[QD * NPAIR * 16];        // 18 fused 4x4 complex gates
  __shared__ float  sC[NQ], sS[NQ];

  for (int i = tid; i < QD * NPAIR * 16; i += 256)
    sU[i] = Uall[(size_t)g * QD * NPAIR * 16 + i];
  if (tid < NQ) {
    float ang = 0.5f * x[(size_t)b * NQ + tid];
    sincosf(ang, &sS[tid], &sC[tid]);
  }
  __syncthreads();

  // RY(x) layer on |0...0> is a product state: materialize directly (phi=id).
  for (int i = tid; i < DIMQ; i += 256) {
    float pr = 1.0f;
#pragma unroll
    for (int j = 0; j < NQ; ++j) {
      int bit = (i >> (NQ - 1 - j)) & 1;
      pr *= bit ? sS[j] : sC[j];
    }
    sAmp[i] = make_float2(pr, 0.0f);
  }
  __syncthreads();

  apply_layer<0>(sAmp, sU + 0 * NPAIR * 16, tid);   // gates see identity perm
  apply_layer<1>(sAmp, sU + 1 * NPAIR * 16, tid);   // gates see ring^1
  apply_layer<2>(sAmp, sU + 2 * NPAIR * 16, tid);   // gates see ring^2

  // Readout through ring^3: out[l] = |amp[phi(l)]|^2, coalesced on l.
  constexpr Perm P3 = ring_pow(3);
  unsigned base = 0;
#pragma unroll
  for (int k = 0; k < 8; ++k)
    base ^= ((tid >> k) & 1) ? P3.col[k] : 0u;

  float* dst = out + (size_t)b * (NGEN * DIMQ) + (size_t)g * DIMQ;
#pragma unroll 1
  for (int s = 0; s < 16; ++s) {
    unsigned i = base ^ ((s & 1) ? P3.col[8]  : 0u)
                      ^ ((s & 2) ? P3.col[9]  : 0u)
                      ^ ((s & 4) ? P3.col[10] : 0u)
                      ^ ((s & 8) ? P3.col[11] : 0u);
    float2 a = sAmp[i];
    dst[s * 256 + tid] = a.x * a.x + a.y * a.y;
  }
}

// ---------------------------------------------------------------------------
// Kernel 3: converted_x = (x @ lin_w^T + b) * TEMP via V_WMMA_F32_16X16X4_F32.
// M=BATCH, N=DIMQ, K=12 -> exactly 3 chained k=4 WMMAs, no padding.
// A 16x4 layout: lanes 0-15 -> M=lane, {v0:K0, v1:K1}; lanes 16-31 -> {K2,K3}.
// B 4x16 mirrors: lanes 0-15 -> N=lane, {v0:K0, v1:K1}; lanes 16-31 -> {K2,K3}.
// C/D: lane l, vgpr v -> M = v + 8*(l>=16), N = l%16.
// ---------------------------------------------------------------------------
__global__ __launch_bounds__(256)
void linear_wmma_kernel(const float* __restrict__ x,
                        const float* __restrict__ w,
                        const float* __restrict__ bias,
                        float* __restrict__ out) {
  const int lane = threadIdx.x & 31;
  const int wave = threadIdx.x >> 5;
  const int half = lane >> 4;     // selects K pair {0,1} vs {2,3}
  const int lr   = lane & 15;
  const int row0 = blockIdx.x * 16;                 // M tile
  const int col0 = (blockIdx.y * 8 + wave) * 16;    // N tile (one per wave)

  v8f c = {0.f, 0.f, 0.f, 0.f, 0.f, 0.f, 0.f, 0.f};
  const float* xr = x + (size_t)(row0 + lr) * NQ;
  const float* wr = w + (size_t)(col0 + lr) * NQ;

#pragma unroll
  for (int kk = 0; kk < NQ; kk += 4) {
    v2f a, bm;
    a.x  = xr[kk + 2 * half];
    a.y  = xr[kk + 2 * half + 1];
    bm.x = wr[kk + 2 * half];
    bm.y = wr[kk + 2 * half + 1];
    c = __builtin_amdgcn_wmma_f32_16x16x4_f32(false, a, false, bm,
                                              (short)0, c, false, false);
  }

  const float bb = bias[col0 + lr];
  float* dst = out + (size_t)(col0 + lr);
#pragma unroll
  for (int v = 0; v < 8; ++v) {
    int m = row0 + v + 8 * half;
    dst[(size_t)m * DIMQ] = (c[v] + bb) * TEMP;
  }
}

// ---------------------------------------------------------------------------
extern "C" void kernel_launch(void* const* d_in, const int* in_sizes, int n_in,
                              void* d_out, int out_size, void* d_ws, size_t ws_size,
                              hipStream_t stream) {
  const float* x  = (const float*)d_in[0];   // (4096, 12)
  const float* qp = (const float*)d_in[1];   // (2, 108)
  const float* lw = (const float*)d_in[2];   // (4096, 12)
  const float* lb = (const float*)d_in[3];   // (4096,)
  // d_in[4] = valid_state_mask (all ones, unused by reference math)

  float* out       = (float*)d_out;
  float* out_probs = out;                                   // (4096, 8192)
  float* out_conv  = out + (size_t)BATCH * NGEN * DIMQ;     // (4096, 4096)
  float2* U        = (float2*)d_ws;                         // 36 fused 4x4 gates

  fuse_pairs_kernel<<<1, 64, 0, stream>>>(qp, U);
  circuit_kernel<<<dim3(BATCH, NGEN), 256, 0, stream>>>(x, U, out_probs);
  linear_wmma_kernel<<<dim3(BATCH / 16, DIMQ / 128), 256, 0, stream>>>(
      x, lw, lb, out_conv);
}